// OneHopSumNodeLabelAggregator_8108898255210
// MI455X (gfx1250) — compile-verified
//
#include <hip/hip_runtime.h>
#include <cstdint>
#include <cstddef>

// ---------------------------------------------------------------------------
// segment_sum(x[src], dst)  ==  adj^T @ x  for a sparse random graph.
// Strategy: build CSR-by-dst in workspace (only 1.6M u32 atomics), then a
// wave32-per-node gather phase that streams 512B rows of x (L2-resident,
// 25.6MB) through the gfx1250 async global->LDS engine, double-buffered with
// s_wait_asynccnt. No f32 atomics anywhere.
// ---------------------------------------------------------------------------

#define THREADS 256
#define WAVES_PER_BLOCK 8   // 256 threads / wave32

// Builtin signature (from clang diagnostic): b128 variant takes
// (v4i addrspace(1)*, v4i addrspace(3)*, imm offset, imm cpol).
typedef __attribute__((__vector_size__(16))) int v4i;
typedef __attribute__((address_space(1))) v4i GAS_v4i;   // global
typedef __attribute__((address_space(3))) v4i LAS_v4i;   // LDS

// Async copy of one 16B chunk per lane: global -> LDS (ASYNCcnt-tracked).
// LDS addrspace(3) pointers are the low 32 bits of the generic address
// (the high bits of a generic LDS address hold the shared aperture base).
__device__ __forceinline__ void async_copy16(const float* g, const float* l) {
#if __has_builtin(__builtin_amdgcn_global_load_async_to_lds_b128)
  __builtin_amdgcn_global_load_async_to_lds_b128(
      (GAS_v4i*)(unsigned long long)(uintptr_t)g,
      (LAS_v4i*)(unsigned int)(uintptr_t)l,
      0, 0);
#elif defined(__AMDGCN__)
  unsigned int lds_off = (unsigned int)(uintptr_t)l;
  unsigned long long ga = (unsigned long long)(uintptr_t)g;
  asm volatile("global_load_async_to_lds_b128 %0, %1, off"
               :: "v"(lds_off), "v"(ga) : "memory");
#endif
}

__device__ __forceinline__ void wait_async_le1() {
#if __has_builtin(__builtin_amdgcn_s_wait_asynccnt)
  __builtin_amdgcn_s_wait_asynccnt(1);
#elif defined(__AMDGCN__)
  asm volatile("s_wait_asynccnt 0x1" ::: "memory");
#endif
}

__device__ __forceinline__ void wait_async_le0() {
#if __has_builtin(__builtin_amdgcn_s_wait_asynccnt)
  __builtin_amdgcn_s_wait_asynccnt(0);
#elif defined(__AMDGCN__)
  asm volatile("s_wait_asynccnt 0x0" ::: "memory");
#endif
}

// ---- Phase 1: zero the (N+1)-entry offsets/count array --------------------
__global__ void k_zero(int* __restrict__ a, int n) {
  int i = blockIdx.x * blockDim.x + threadIdx.x;
  if (i < n) a[i] = 0;
}

// ---- Phase 2: per-dst degree histogram (u32 atomics, L2) ------------------
__global__ void k_count(const int* __restrict__ dst, int* __restrict__ cnt, int E) {
  int e = blockIdx.x * blockDim.x + threadIdx.x;
  if (e < E) atomicAdd(&cnt[dst[e]], 1);
}

// ---- Phase 3: exclusive scan in place, single 1024-thread workgroup -------
// 50K elements => 49 iterations of a 1024-wide LDS Hillis-Steele scan.
__global__ void k_scan(int* __restrict__ off, int n) {
  __shared__ int tmp[1024];
  const int t = threadIdx.x;
  int carry = 0;
  for (int base = 0; base < n; base += 1024) {
    int i = base + t;
    int v = (i < n) ? off[i] : 0;
    tmp[t] = v;
    __syncthreads();
    for (int d = 1; d < 1024; d <<= 1) {
      int addend = (t >= d) ? tmp[t - d] : 0;
      __syncthreads();
      tmp[t] += addend;
      __syncthreads();
    }
    if (i < n) off[i] = carry + (tmp[t] - v);  // exclusive + carry
    carry += tmp[1023];
    __syncthreads();
  }
  if (t == 0) off[n] = carry;  // total == E
}

// ---- Phase 4: CSR fill. Atomic cursor reuses off[]; afterwards ------------
//      off[d] == end offset of node d (== original start of d+1).
__global__ void k_fill(const int* __restrict__ src, const int* __restrict__ dst,
                       int* __restrict__ cursor, int* __restrict__ esrc, int E) {
  int e = blockIdx.x * blockDim.x + threadIdx.x;
  if (e < E) {
    int pos = atomicAdd(&cursor[dst[e]], 1);
    esrc[pos] = src[e];
  }
}

// ---- Phase 5: wave-per-node gather-sum with async double buffering --------
__global__ void __launch_bounds__(THREADS)
k_gather(const float* __restrict__ x, const int* __restrict__ off,
         const int* __restrict__ esrc, float* __restrict__ out, int n) {
  __shared__ float lbuf[WAVES_PER_BLOCK][2][128] __attribute__((aligned(16)));
  const int wave = threadIdx.x >> 5;   // wave32 on gfx1250
  const int lane = threadIdx.x & 31;
  const int node = blockIdx.x * WAVES_PER_BLOCK + wave;
  if (node >= n) return;               // wave-uniform: EXEC stays all-ones

  const int start = (node == 0) ? 0 : off[node - 1];
  const int end   = off[node];
  const int nb    = end - start;

  float a0 = 0.f, a1 = 0.f, a2 = 0.f, a3 = 0.f;

  if (nb > 0) {
    // Prime buffer 0 with the first row. Each lane fetches 16B; the wave's
    // 32 lanes cover one full 512B feature row of x.
    int s0 = esrc[start];
    async_copy16(x + (size_t)s0 * 128 + lane * 4, &lbuf[wave][0][lane * 4]);
    for (int i = 0; i < nb; ++i) {
      const int cur = i & 1;
      if (i + 1 < nb) {
        int s1 = esrc[start + i + 1];
        async_copy16(x + (size_t)s1 * 128 + lane * 4,
                     &lbuf[wave][cur ^ 1][lane * 4]);
        wait_async_le1();  // async loads complete in order: row i is ready
      } else {
        wait_async_le0();
      }
      const float4 v = *(const float4*)&lbuf[wave][cur][lane * 4];
      a0 += v.x; a1 += v.y; a2 += v.z; a3 += v.w;
    }
  }

  float4 r; r.x = a0; r.y = a1; r.z = a2; r.w = a3;
  *(float4*)(out + (size_t)node * 128 + lane * 4) = r;  // coalesced b128
}

// ---------------------------------------------------------------------------
extern "C" void kernel_launch(void* const* d_in, const int* in_sizes, int n_in,
                              void* d_out, int out_size, void* d_ws, size_t ws_size,
                              hipStream_t stream) {
  (void)n_in; (void)ws_size; (void)out_size;
  const float* x     = (const float*)d_in[0];
  const int*   edges = (const int*)d_in[1];      // (2, E) flat: row0=src, row1=dst
  float*       out   = (float*)d_out;

  const int N = in_sizes[0] / 128;               // 50000
  const int E = in_sizes[1] / 2;                 // 800000
  if (N <= 0 || E <= 0) return;

  const int* src = edges;
  const int* dst = edges + E;

  // Workspace layout: [ off: (N+1) ints ][ esrc: E ints ]  (~3.4 MB)
  int* off  = (int*)d_ws;
  int* esrc = off + (N + 1);

  k_zero <<<(N + 1 + THREADS - 1) / THREADS, THREADS, 0, stream>>>(off, N + 1);
  k_count<<<(E + THREADS - 1) / THREADS,     THREADS, 0, stream>>>(dst, off, E);
  k_scan <<<1, 1024, 0, stream>>>(off, N);
  k_fill <<<(E + THREADS - 1) / THREADS,     THREADS, 0, stream>>>(src, dst, off, esrc, E);
  k_gather<<<(N + WAVES_PER_BLOCK - 1) / WAVES_PER_BLOCK, THREADS, 0, stream>>>(
      x, off, esrc, out, N);
}